// GAT_87144886436013
// MI455X (gfx1250) — compile-verified
//
#include <hip/hip_runtime.h>
#include <math.h>

#define HEADS 4
#define HID 32
#define FD 128   // HEADS*HID == IN == 128

typedef __attribute__((ext_vector_type(2))) float v2f;
typedef __attribute__((ext_vector_type(4))) float v4f;
typedef __attribute__((ext_vector_type(8))) float v8f;

__device__ __forceinline__ float elu1(float x) { return x > 0.f ? x : (expf(x) - 1.f); }
__device__ __forceinline__ float lrelu(float x) { return x > 0.f ? x : 0.2f * x; }

// Order-preserving float <-> uint encoding for atomicMax on floats
__device__ __forceinline__ unsigned fenc(float f) {
    unsigned u = __float_as_uint(f);
    return (u & 0x80000000u) ? ~u : (u | 0x80000000u);
}
__device__ __forceinline__ float fdec(unsigned u) {
    return __uint_as_float((u & 0x80000000u) ? (u & 0x7fffffffu) : ~u);
}

// ---------------------------------------------------------------------------
// Pure WMMA f32 GEMM: D[m, n] = A[m, 0:128] @ B[0:128, n] (+ cinit[n])
// One wave per 16x16 tile, K = 128 via 32x V_WMMA_F32_16X16X4_F32.
// Inner loop is branch-free: one b64 A load + two B loads + one WMMA.
// Requirements (guaranteed by launcher): nrows % 16 == 0; B/cinit valid for
// all 16 tile columns (pre-padded). Only the D store is column-guarded.
// ---------------------------------------------------------------------------
__global__ void gat_gemm_wmma(const float* __restrict__ A,
                              const float* __restrict__ B, int ldB,
                              const float* __restrict__ cinit,
                              float* __restrict__ D, int ldD, int ncolsD,
                              int nrows)
{
    const int lane = threadIdx.x & 31;
    const int half = lane >> 4;   // 0: K pair {0,1}, 1: K pair {2,3}
    const int l16  = lane & 15;
    const int m0 = blockIdx.x * 16;
    const int n0 = blockIdx.y * 16;
    if (m0 >= nrows) return;

    const int row = m0 + l16;     // A-matrix row for this lane
    const int col = n0 + l16;     // B/C/D column for this lane

    v8f acc;
    const float ci = (cinit != nullptr) ? cinit[col] : 0.f;
#pragma unroll
    for (int g = 0; g < 8; ++g) acc[g] = ci;

    const float* arow = A + (size_t)row * FD;
    const float* bcol = B + col;

#pragma unroll 8
    for (int k0 = 0; k0 < FD; k0 += 4) {
        const int kk = k0 + 2 * half;
        // A fragment (16x4 f32): lanes 0-15 -> K={k0,k0+1}; lanes 16-31 -> K={k0+2,k0+3}
        v2f av = *(const v2f*)(arow + kk);
        // B fragment (4x16 f32): lane -> column l16, same K split as A
        v2f bv;
        bv.x = bcol[(size_t)kk * ldB];
        bv.y = bcol[(size_t)(kk + 1) * ldB];
        acc = __builtin_amdgcn_wmma_f32_16x16x4_f32(
            /*neg_a=*/false, av, /*neg_b=*/false, bv,
            /*c_mod=*/(short)0, acc, /*reuse_a=*/false, /*reuse_b=*/false);
    }

    // C/D layout: VGPR g -> row m0 + g + 8*half, column n0 + l16.
    // Single per-lane predicate; 8 stores with constant stride ldD.
    if (col < ncolsD) {
        float* dptr = D + (size_t)(m0 + 8 * half) * ldD + col;
#pragma unroll
        for (int g = 0; g < 8; ++g) dptr[(size_t)g * ldD] = acc[g];
    }
}

// ---------------------------------------------------------------------------
// In-place elementwise: buf[i] += bias[i % 128]; optionally ELU.
// ---------------------------------------------------------------------------
__global__ void gat_bias_act(float* __restrict__ buf, const float* __restrict__ bias,
                             int total, int do_elu)
{
    int i = blockIdx.x * blockDim.x + threadIdx.x;
    if (i >= total) return;
    float v = buf[i] + bias[i & (FD - 1)];
    if (do_elu) v = elu1(v);
    buf[i] = v;
}

// ---------------------------------------------------------------------------
// Zero-pad Wout (128x8 -> 128x16) and bout (8 -> 16) for the output GEMM.
// ---------------------------------------------------------------------------
__global__ void gat_pad_wout(const float* __restrict__ Wout, const float* __restrict__ bout,
                             float* __restrict__ Wpad, float* __restrict__ cpad)
{
    int i = blockIdx.x * blockDim.x + threadIdx.x;
    if (i < FD * 16) {
        int k = i >> 4, c = i & 15;
        Wpad[i] = (c < 8) ? Wout[k * 8 + c] : 0.f;
    }
    if (i < 16) cpad[i] = (i < 8) ? bout[i] : 0.f;
}

// ---------------------------------------------------------------------------
// Per-(node, head) attention scores: a_s = <h, att_src>, a_d = <h, att_dst>
// ---------------------------------------------------------------------------
__global__ void gat_scores(const float* __restrict__ h,
                           const float* __restrict__ att_s,
                           const float* __restrict__ att_d,
                           float* __restrict__ a_s, float* __restrict__ a_d, int Nn)
{
    int t = blockIdx.x * blockDim.x + threadIdx.x;
    if (t >= Nn * HEADS) return;
    int n = t >> 2, hd = t & 3;
    const float* hp = h + (size_t)n * FD + hd * HID;
    const float* sp = att_s + hd * HID;
    const float* dp = att_d + hd * HID;
    float ss = 0.f, dd = 0.f;
#pragma unroll
    for (int c = 0; c < HID; ++c) {
        float hv = hp[c];
        ss += hv * sp[c];
        dd += hv * dp[c];
    }
    a_s[t] = ss;
    a_d[t] = dd;
}

// ---------------------------------------------------------------------------
// Edge pass 1: per-destination max of leaky_relu logits (segment_max)
// ---------------------------------------------------------------------------
__global__ void gat_edge_max(const int* __restrict__ src, const int* __restrict__ dst,
                             const float* __restrict__ a_s, const float* __restrict__ a_d,
                             unsigned* __restrict__ m_enc, int E, int Nn)
{
    int e = blockIdx.x * blockDim.x + threadIdx.x;
    if (e >= E + Nn) return;
    int s, d;
    if (e < E) { s = src[e]; d = dst[e]; } else { s = e - E; d = s; }
    v4f vs = *(const v4f*)(a_s + (size_t)s * HEADS);   // b128 load
    v4f vd = *(const v4f*)(a_d + (size_t)d * HEADS);   // b128 load
#pragma unroll
    for (int hd = 0; hd < HEADS; ++hd) {
        float v = lrelu(vs[hd] + vd[hd]);
        atomicMax(&m_enc[d * HEADS + hd], fenc(v));
    }
}

// ---------------------------------------------------------------------------
// Edge pass 2: denom[dst] += exp(logit - max[dst])  (segment_sum)
// ---------------------------------------------------------------------------
__global__ void gat_edge_sum(const int* __restrict__ src, const int* __restrict__ dst,
                             const float* __restrict__ a_s, const float* __restrict__ a_d,
                             const unsigned* __restrict__ m_enc,
                             float* __restrict__ denom, int E, int Nn)
{
    int e = blockIdx.x * blockDim.x + threadIdx.x;
    if (e >= E + Nn) return;
    int s, d;
    if (e < E) { s = src[e]; d = dst[e]; } else { s = e - E; d = s; }
    v4f vs = *(const v4f*)(a_s + (size_t)s * HEADS);
    v4f vd = *(const v4f*)(a_d + (size_t)d * HEADS);
#pragma unroll
    for (int hd = 0; hd < HEADS; ++hd) {
        float v = lrelu(vs[hd] + vd[hd]);
        float m = fdec(m_enc[d * HEADS + hd]);
        atomicAdd(&denom[d * HEADS + hd], expf(v - m));
    }
}

// ---------------------------------------------------------------------------
// Edge pass 3: one wave32 per (edge, head); lane = channel within head.
// agg[dst, head, c] += h[src, head, c] * alpha(edge, head)
// ---------------------------------------------------------------------------
__global__ void gat_edge_msg(const int* __restrict__ src, const int* __restrict__ dst,
                             const float* __restrict__ h,
                             const float* __restrict__ a_s, const float* __restrict__ a_d,
                             const unsigned* __restrict__ m_enc,
                             const float* __restrict__ denom,
                             float* __restrict__ agg, int E, int Nn)
{
    int gid = blockIdx.x * blockDim.x + threadIdx.x;
    int wid = gid >> 5;                 // global wave id = edge*HEADS + head
    if (wid >= (E + Nn) * HEADS) return;
    int lane = threadIdx.x & 31;
    int e  = wid >> 2;
    int hd = wid & 3;
    int s, d;
    if (e < E) { s = src[e]; d = dst[e]; } else { s = e - E; d = s; }

    // uniform-address loads across the wave -> hardware broadcast
    float v = lrelu(a_s[s * HEADS + hd] + a_d[d * HEADS + hd]);
    float m = fdec(m_enc[d * HEADS + hd]);
    float t = expf(v - m);
    float alpha = t / (denom[d * HEADS + hd] + 1e-16f);

    int c = hd * HID + lane;
    float val = h[(size_t)s * FD + c] * alpha;
    atomicAdd(&agg[(size_t)d * FD + c], val);
}

// ---------------------------------------------------------------------------
// Host launcher
// ---------------------------------------------------------------------------
extern "C" void kernel_launch(void* const* d_in, const int* in_sizes, int n_in,
                              void* d_out, int out_size, void* d_ws, size_t ws_size,
                              hipStream_t stream)
{
    const float* x    = (const float*)d_in[0];
    const int*   ei   = (const int*)d_in[1];
    const float* W1   = (const float*)d_in[2];
    const float* as1  = (const float*)d_in[3];
    const float* ad1  = (const float*)d_in[4];
    const float* b1   = (const float*)d_in[5];
    const float* W2   = (const float*)d_in[6];
    const float* as2  = (const float*)d_in[7];
    const float* ad2  = (const float*)d_in[8];
    const float* b2   = (const float*)d_in[9];
    const float* Wout = (const float*)d_in[10];
    const float* bout = (const float*)d_in[11];
    float* out = (float*)d_out;

    const int Nn = in_sizes[0] / FD;   // 50000 (multiple of 16)
    const int E  = in_sizes[1] / 2;    // 1600000
    const int ET = E + Nn;             // edges incl. self-loops

    const int* srcI = ei;
    const int* dstI = ei + E;

    // Workspace layout (floats)
    float* ws       = (float*)d_ws;
    float* hbuf     = ws;                                   // N*128
    float* aggbuf   = hbuf   + (size_t)Nn * FD;             // N*128
    float* a_s      = aggbuf + (size_t)Nn * FD;             // N*4
    float* a_d      = a_s    + (size_t)Nn * HEADS;          // N*4
    float* denom    = a_d    + (size_t)Nn * HEADS;          // N*4
    unsigned* m_enc = (unsigned*)(denom + (size_t)Nn * HEADS); // N*4
    float* Wpad     = (float*)(m_enc + (size_t)Nn * HEADS); // 128*16
    float* cpad     = Wpad + FD * 16;                       // 16

    const size_t nhBytes = (size_t)Nn * HEADS * sizeof(float);
    const size_t nfBytes = (size_t)Nn * FD * sizeof(float);
    const int totalNF = Nn * FD;

    const dim3 gGemm((Nn + 15) / 16, FD / 16);
    const int thrScore = 256, blkScore = (Nn * HEADS + thrScore - 1) / thrScore;
    const int thrEdge = 256, blkEdge = (ET + thrEdge - 1) / thrEdge;
    const int blkAct = (totalNF + 255) / 256;
    const long long msgThreads = (long long)ET * HEADS * 32;
    const int blkMsg = (int)((msgThreads + 255) / 256);

    // Prepare padded output weights (deterministic, every call)
    gat_pad_wout<<<(FD * 16 + 255) / 256, 256, 0, stream>>>(Wout, bout, Wpad, cpad);

    // ---------------- Layer 1: h1 = x @ W1 ----------------
    gat_gemm_wmma<<<gGemm, 32, 0, stream>>>(x, W1, FD, nullptr, hbuf, FD, FD, Nn);
    hipMemsetAsync(m_enc, 0, nhBytes, stream);
    hipMemsetAsync(denom, 0, nhBytes, stream);
    hipMemsetAsync(aggbuf, 0, nfBytes, stream);
    gat_scores<<<blkScore, thrScore, 0, stream>>>(hbuf, as1, ad1, a_s, a_d, Nn);
    gat_edge_max<<<blkEdge, thrEdge, 0, stream>>>(srcI, dstI, a_s, a_d, m_enc, E, Nn);
    gat_edge_sum<<<blkEdge, thrEdge, 0, stream>>>(srcI, dstI, a_s, a_d, m_enc, denom, E, Nn);
    gat_edge_msg<<<blkMsg, 256, 0, stream>>>(srcI, dstI, hbuf, a_s, a_d, m_enc, denom, aggbuf, E, Nn);

    // ---------------- Layer 2: h2 = elu(agg1 + b1) @ W2 ----------------
    gat_bias_act<<<blkAct, 256, 0, stream>>>(aggbuf, b1, totalNF, /*do_elu=*/1);
    gat_gemm_wmma<<<gGemm, 32, 0, stream>>>(aggbuf, W2, FD, nullptr, hbuf, FD, FD, Nn);
    hipMemsetAsync(m_enc, 0, nhBytes, stream);
    hipMemsetAsync(denom, 0, nhBytes, stream);
    hipMemsetAsync(aggbuf, 0, nfBytes, stream);
    gat_scores<<<blkScore, thrScore, 0, stream>>>(hbuf, as2, ad2, a_s, a_d, Nn);
    gat_edge_max<<<blkEdge, thrEdge, 0, stream>>>(srcI, dstI, a_s, a_d, m_enc, E, Nn);
    gat_edge_sum<<<blkEdge, thrEdge, 0, stream>>>(srcI, dstI, a_s, a_d, m_enc, denom, E, Nn);
    gat_edge_msg<<<blkMsg, 256, 0, stream>>>(srcI, dstI, hbuf, a_s, a_d, m_enc, denom, aggbuf, E, Nn);

    // ---------------- Output: (agg2 + b2) @ Wout + bout ----------------
    gat_bias_act<<<blkAct, 256, 0, stream>>>(aggbuf, b2, totalNF, /*do_elu=*/0);
    const dim3 gOut((Nn + 15) / 16, 1);
    gat_gemm_wmma<<<gOut, 32, 0, stream>>>(aggbuf, Wpad, 16, cpad, out, 8, 8, Nn);
}